// SetAbstraction_4501125726462
// MI455X (gfx1250) — compile-verified
//
#include <hip/hip_runtime.h>

typedef float v2f __attribute__((ext_vector_type(2)));
typedef float v8f __attribute__((ext_vector_type(8)));

#define BB 16
#define NN 4096
#define MM 1024
#define KNbr 32
#define NROWS (BB*MM*KNbr)   // 524288
#define TAUf 10.0f
#define BN_EPSf 1e-5f

__device__ __forceinline__ v8f wmma_f32(v2f a, v2f b, v8f c) {
  // D = A(16x4) * B(4x16) + C, fp32, wave32
  return __builtin_amdgcn_wmma_f32_16x16x4_f32(false, a, false, b, (short)0, c, false, false);
}

// ---------------------------------------------------------------------------
// 1) Farthest point sampling: one block per batch. Writes idx (B,M) and
//    centroids (B,M,3) into d_out's first region.
// ---------------------------------------------------------------------------
__global__ __launch_bounds__(256) void fps_kernel(const float* __restrict__ xyz,
                                                  int* __restrict__ idx,
                                                  float* __restrict__ centroids) {
  __shared__ float sx[NN], sy[NN], sz[NN];
  __shared__ int s_idx[MM];
  __shared__ int s_last;
  __shared__ float wval[8];
  __shared__ int widx[8];

  const int b = blockIdx.x;
  const int t = threadIdx.x;
  const float* xb = xyz + (size_t)b * NN * 3;
  for (int p = t; p < NN; p += 256) {
    sx[p] = xb[p * 3 + 0];
    sy[p] = xb[p * 3 + 1];
    sz[p] = xb[p * 3 + 2];
  }
  float dmin[16];
#pragma unroll
  for (int j = 0; j < 16; ++j) dmin[j] = 1e10f;
  if (t == 0) { s_last = 0; s_idx[0] = 0; }
  int* idxb = idx + b * MM;
  if (t == 0) idxb[0] = 0;
  __syncthreads();

  for (int it = 1; it < MM; ++it) {
    const int last = s_last;
    const float lx = sx[last], ly = sy[last], lz = sz[last];
    float bestv = -1.0f;
    int besti = 0;
#pragma unroll
    for (int j = 0; j < 16; ++j) {
      const int p = t * 16 + j;
      const float dx = sx[p] - lx, dy = sy[p] - ly, dz = sz[p] - lz;
      const float d = dx * dx + dy * dy + dz * dz;
      const float dm = fminf(dmin[j], d);
      dmin[j] = dm;
      if (dm > bestv) { bestv = dm; besti = p; }
    }
    // wave32 reduction, ties -> lower index (thread/lane order is index order)
    for (int off = 16; off > 0; off >>= 1) {
      const float ov = __shfl_down(bestv, off);
      const int oi = __shfl_down(besti, off);
      if (ov > bestv || (ov == bestv && oi < besti)) { bestv = ov; besti = oi; }
    }
    if ((t & 31) == 0) { wval[t >> 5] = bestv; widx[t >> 5] = besti; }
    __syncthreads();
    if (t == 0) {
      float bv = wval[0]; int bi = widx[0];
      for (int w = 1; w < 8; ++w)
        if (wval[w] > bv || (wval[w] == bv && widx[w] < bi)) { bv = wval[w]; bi = widx[w]; }
      s_last = bi; s_idx[it] = bi; idxb[it] = bi;
    }
    __syncthreads();
  }
  for (int m = t; m < MM; m += 256) {
    const int p = s_idx[m];
    float* co = centroids + ((size_t)b * MM + m) * 3;
    co[0] = sx[p]; co[1] = sy[p]; co[2] = sz[p];
  }
}

// ---------------------------------------------------------------------------
// 2) KNN via 4-pass radix select on float bit patterns (d2 >= 0 so uint order
//    == float order). One block per (b,m). Output indices ascending (the
//    downstream pipeline is permutation-invariant over K).
// ---------------------------------------------------------------------------
__global__ __launch_bounds__(256) void knn_kernel(const float* __restrict__ xyz,
                                                  const int* __restrict__ idx,
                                                  int* __restrict__ gidx) {
  __shared__ unsigned sd[NN];
  __shared__ unsigned hist[256];
  __shared__ unsigned s_prefix, s_kth, s_cbefore, s_cntLess, s_cntTie;
  __shared__ int s_out[KNbr];

  const int bm = blockIdx.x;
  const int b = bm >> 10;
  const int t = threadIdx.x;
  const float* xb = xyz + (size_t)b * NN * 3;
  const int ci = idx[bm];
  const float cx = xb[ci * 3 + 0], cy = xb[ci * 3 + 1], cz = xb[ci * 3 + 2];
  for (int p = t; p < NN; p += 256) {
    const float dx = xb[p * 3 + 0] - cx, dy = xb[p * 3 + 1] - cy, dz = xb[p * 3 + 2] - cz;
    sd[p] = __float_as_uint(dx * dx + dy * dy + dz * dz);
  }
  if (t == 0) { s_prefix = 0u; s_kth = KNbr; s_cbefore = 0u; s_cntLess = 0u; s_cntTie = 0u; }
  __syncthreads();

  for (int pass = 0; pass < 4; ++pass) {
    const int shift = 24 - pass * 8;
    hist[t] = 0u;
    __syncthreads();
    const unsigned prefix = s_prefix;
    const unsigned pmask = (pass == 0) ? 0u : (0xFFFFFFFFu << (32 - 8 * pass));
    for (int p = t; p < NN; p += 256) {
      const unsigned v = sd[p];
      if ((v & pmask) == prefix) atomicAdd(&hist[(v >> shift) & 0xFFu], 1u);
    }
    __syncthreads();
    if (t == 0) {
      unsigned kth = s_kth, cum = 0u;
      int sel = 0;
      for (int bin = 0; bin < 256; ++bin) {
        const unsigned c = hist[bin];
        if (cum + c >= kth) { sel = bin; s_kth = kth - cum; s_cbefore += cum; break; }
        cum += c;
      }
      s_prefix = prefix | ((unsigned)sel << shift);
    }
    __syncthreads();
  }
  const unsigned T = s_prefix;        // value of rank K (1-indexed)
  const unsigned cless = s_cbefore;   // count strictly < T
  const unsigned tieneed = KNbr - cless;
  for (int p = t; p < NN; p += 256) {
    const unsigned v = sd[p];
    if (v < T) {
      const unsigned pos = atomicAdd(&s_cntLess, 1u);
      s_out[pos] = p;
    } else if (v == T) {
      const unsigned q = atomicAdd(&s_cntTie, 1u);
      if (q < tieneed) s_out[cless + q] = p;
    }
  }
  __syncthreads();
  if (t < KNbr) {  // sort ascending by index -> deterministic order
    const int val = s_out[t];
    int rank = 0;
    for (int j = 0; j < KNbr; ++j) rank += (s_out[j] < val);
    gidx[(size_t)bm * KNbr + rank] = val;
  }
}

// ---------------------------------------------------------------------------
// GEMM layer 1: gather(xyz-rel ++ features) (32 x 67, pad 68) * W1^T (67x64)
// using V_WMMA_F32_16X16X4_F32. One block per (b,m); 8 waves = 2x4 tiles of
// 16x16. Also emits per-block channel sum/sumsq partials for BN1.
// ---------------------------------------------------------------------------
__global__ __launch_bounds__(256) void gemm1_kernel(
    const float* __restrict__ xyz, const float* __restrict__ feat,
    const float* __restrict__ centroids, const int* __restrict__ gidx,
    const float* __restrict__ W1, float* __restrict__ Ypre,
    float* __restrict__ psum, float* __restrict__ psq) {
  __shared__ float Xl[32][68];
  __shared__ float Wl[64][68];
  __shared__ int sg[32];
  __shared__ float scent[3];
  __shared__ float part_s[4][64];
  __shared__ float part_q[4][64];

  const int bm = blockIdx.x;
  const int b = bm >> 10;
  const int t = threadIdx.x;

  if (t < 32) sg[t] = gidx[(size_t)bm * KNbr + t];
  if (t < 3) scent[t] = centroids[(size_t)bm * 3 + t];
  for (int e = t; e < 64 * 68; e += 256) {
    const int o = e / 68, c = e % 68;
    Wl[o][c] = (c < 67) ? W1[o * 67 + c] : 0.0f;
  }
  __syncthreads();
  for (int e = t; e < 32 * 68; e += 256) {
    const int kk = e / 68, c = e % 68;
    const int g = sg[kk];
    float v;
    if (c < 3)       v = xyz[((size_t)b * NN + g) * 3 + c] - scent[c];
    else if (c < 67) v = feat[((size_t)b * NN + g) * 64 + (c - 3)];
    else             v = 0.0f;
    Xl[kk][c] = v;
  }
  __syncthreads();

  const int w = t >> 5, lane = t & 31;
  const int n = lane & 15, kh = lane >> 4;
  const int rt = w & 1, ct = w >> 1;
  v8f acc = {0, 0, 0, 0, 0, 0, 0, 0};
#pragma unroll
  for (int ks = 0; ks < 17; ++ks) {
    const int k0 = ks * 4 + 2 * kh;
    const v2f a = *(const v2f*)&Xl[rt * 16 + n][k0];
    const v2f bf = *(const v2f*)&Wl[ct * 16 + n][k0];
    acc = wmma_f32(a, bf, acc);
  }
  float s = 0.0f, q = 0.0f;
  const size_t rowbase = (size_t)bm * 32 + rt * 16 + 8 * kh;
  const int ch = ct * 16 + n;
#pragma unroll
  for (int v = 0; v < 8; ++v) {
    const float y = acc[v];
    Ypre[(rowbase + v) * 64 + ch] = y;
    s += y; q += y * y;
  }
  part_s[rt * 2 + kh][ch] = s;
  part_q[rt * 2 + kh][ch] = q;
  __syncthreads();
  if (t < 64) {
    psum[(size_t)bm * 64 + t] = part_s[0][t] + part_s[1][t] + part_s[2][t] + part_s[3][t];
    psq [(size_t)bm * 64 + t] = part_q[0][t] + part_q[1][t] + part_q[2][t] + part_q[3][t];
  }
}

// ---------------------------------------------------------------------------
// GEMM layer 2: BN1+ReLU fused on load, (32 x 64) * W2^T (64x64).
// ---------------------------------------------------------------------------
__global__ __launch_bounds__(256) void gemm2_kernel(
    const float* __restrict__ Yin, const float* __restrict__ gsum,
    const float* __restrict__ gsq, const float* __restrict__ gamma,
    const float* __restrict__ beta, const float* __restrict__ W,
    float* __restrict__ Ypre, float* __restrict__ psum, float* __restrict__ psq) {
  __shared__ float Xl[32][68];
  __shared__ float Wl[64][68];
  __shared__ float sca[64], shf[64];
  __shared__ float part_s[4][64];
  __shared__ float part_q[4][64];

  const int bm = blockIdx.x;
  const int t = threadIdx.x;
  if (t < 64) {
    const float mu = gsum[t] * (1.0f / NROWS);
    const float var = gsq[t] * (1.0f / NROWS) - mu * mu;
    const float a = gamma[t] * rsqrtf(var + BN_EPSf);
    sca[t] = a; shf[t] = beta[t] - mu * a;
  }
  for (int e = t; e < 64 * 68; e += 256) {
    const int o = e / 68, c = e % 68;
    Wl[o][c] = (c < 64) ? W[o * 64 + c] : 0.0f;
  }
  __syncthreads();
  for (int e = t; e < 32 * 64; e += 256) {
    const int kk = e >> 6, c = e & 63;
    const float v = Yin[((size_t)bm * 32 + kk) * 64 + c] * sca[c] + shf[c];
    Xl[kk][c] = fmaxf(v, 0.0f);
  }
  __syncthreads();

  const int w = t >> 5, lane = t & 31;
  const int n = lane & 15, kh = lane >> 4;
  const int rt = w & 1, ct = w >> 1;
  v8f acc = {0, 0, 0, 0, 0, 0, 0, 0};
#pragma unroll
  for (int ks = 0; ks < 16; ++ks) {
    const int k0 = ks * 4 + 2 * kh;
    const v2f a = *(const v2f*)&Xl[rt * 16 + n][k0];
    const v2f bf = *(const v2f*)&Wl[ct * 16 + n][k0];
    acc = wmma_f32(a, bf, acc);
  }
  float s = 0.0f, q = 0.0f;
  const size_t rowbase = (size_t)bm * 32 + rt * 16 + 8 * kh;
  const int ch = ct * 16 + n;
#pragma unroll
  for (int v = 0; v < 8; ++v) {
    const float y = acc[v];
    Ypre[(rowbase + v) * 64 + ch] = y;
    s += y; q += y * y;
  }
  part_s[rt * 2 + kh][ch] = s;
  part_q[rt * 2 + kh][ch] = q;
  __syncthreads();
  if (t < 64) {
    psum[(size_t)bm * 64 + t] = part_s[0][t] + part_s[1][t] + part_s[2][t] + part_s[3][t];
    psq [(size_t)bm * 64 + t] = part_q[0][t] + part_q[1][t] + part_q[2][t] + part_q[3][t];
  }
}

// ---------------------------------------------------------------------------
// GEMM layer 3: BN2+ReLU fused on load, (32 x 64) * W3^T (64x128).
// 8 waves, each covers one 16-col tile for both row tiles (2 accumulators).
// ---------------------------------------------------------------------------
__global__ __launch_bounds__(256) void gemm3_kernel(
    const float* __restrict__ Yin, const float* __restrict__ gsum,
    const float* __restrict__ gsq, const float* __restrict__ gamma,
    const float* __restrict__ beta, const float* __restrict__ W,
    float* __restrict__ Ypre, float* __restrict__ psum, float* __restrict__ psq) {
  __shared__ float Xl[32][68];
  __shared__ float Wl[128][68];
  __shared__ float sca[64], shf[64];
  __shared__ float part_s[2][128];
  __shared__ float part_q[2][128];

  const int bm = blockIdx.x;
  const int t = threadIdx.x;
  if (t < 64) {
    const float mu = gsum[t] * (1.0f / NROWS);
    const float var = gsq[t] * (1.0f / NROWS) - mu * mu;
    const float a = gamma[t] * rsqrtf(var + BN_EPSf);
    sca[t] = a; shf[t] = beta[t] - mu * a;
  }
  for (int e = t; e < 128 * 68; e += 256) {
    const int o = e / 68, c = e % 68;
    Wl[o][c] = (c < 64) ? W[o * 64 + c] : 0.0f;
  }
  __syncthreads();
  for (int e = t; e < 32 * 64; e += 256) {
    const int kk = e >> 6, c = e & 63;
    const float v = Yin[((size_t)bm * 32 + kk) * 64 + c] * sca[c] + shf[c];
    Xl[kk][c] = fmaxf(v, 0.0f);
  }
  __syncthreads();

  const int ct = t >> 5, lane = t & 31;
  const int n = lane & 15, kh = lane >> 4;
  v8f acc0 = {0, 0, 0, 0, 0, 0, 0, 0};
  v8f acc1 = {0, 0, 0, 0, 0, 0, 0, 0};
#pragma unroll
  for (int ks = 0; ks < 16; ++ks) {
    const int k0 = ks * 4 + 2 * kh;
    const v2f bf = *(const v2f*)&Wl[ct * 16 + n][k0];
    const v2f a0 = *(const v2f*)&Xl[n][k0];
    const v2f a1 = *(const v2f*)&Xl[16 + n][k0];
    acc0 = wmma_f32(a0, bf, acc0);
    acc1 = wmma_f32(a1, bf, acc1);
  }
  float s = 0.0f, q = 0.0f;
  const int ch = ct * 16 + n;
  const size_t r0 = (size_t)bm * 32 + 8 * kh;
#pragma unroll
  for (int v = 0; v < 8; ++v) {
    const float y0 = acc0[v];
    const float y1 = acc1[v];
    Ypre[(r0 + v) * 128 + ch] = y0;
    Ypre[(r0 + 16 + v) * 128 + ch] = y1;
    s += y0 + y1; q += y0 * y0 + y1 * y1;
  }
  part_s[kh][ch] = s;
  part_q[kh][ch] = q;
  __syncthreads();
  if (t < 128) {
    psum[(size_t)bm * 128 + t] = part_s[0][t] + part_s[1][t];
    psq [(size_t)bm * 128 + t] = part_q[0][t] + part_q[1][t];
  }
}

// ---------------------------------------------------------------------------
// Deterministic partial-sum fold: out[c] = sum over nblk of p[i*nch + c].
// ---------------------------------------------------------------------------
__global__ __launch_bounds__(256) void reduce_kernel(const float* __restrict__ p,
                                                     float* __restrict__ o,
                                                     int nblk, int nch) {
  const int c = blockIdx.x;
  const int t = threadIdx.x;
  float s = 0.0f;
  for (int i = t; i < nblk; i += 256) s += p[(size_t)i * nch + c];
  __shared__ float r[256];
  r[t] = s;
  __syncthreads();
  for (int off = 128; off > 0; off >>= 1) {
    if (t < off) r[t] += r[t + off];
    __syncthreads();
  }
  if (t == 0) o[c] = r[0];
}

// ---------------------------------------------------------------------------
// BN3 + ReLU + LSE pool over K. One block per (b,m), 128 threads (one/chan).
// ---------------------------------------------------------------------------
__global__ __launch_bounds__(128) void lse_kernel(const float* __restrict__ Ypre,
                                                  const float* __restrict__ gsum,
                                                  const float* __restrict__ gsq,
                                                  const float* __restrict__ gamma,
                                                  const float* __restrict__ beta,
                                                  float* __restrict__ out) {
  const int bm = blockIdx.x;
  const int c = threadIdx.x;
  const float mu = gsum[c] * (1.0f / NROWS);
  const float var = gsq[c] * (1.0f / NROWS) - mu * mu;
  const float a = gamma[c] * rsqrtf(var + BN_EPSf);
  const float sh = beta[c] - mu * a;
  float vals[KNbr];
  float vmax = -1e30f;
#pragma unroll
  for (int k = 0; k < KNbr; ++k) {
    const float v = fmaxf(Ypre[((size_t)bm * KNbr + k) * 128 + c] * a + sh, 0.0f);
    vals[k] = v;
    vmax = fmaxf(vmax, v);
  }
  float s = 0.0f;
#pragma unroll
  for (int k = 0; k < KNbr; ++k) s += expf(TAUf * (vals[k] - vmax));
  out[(size_t)bm * 128 + c] = vmax + logf(s) * (1.0f / TAUf);
}

// ---------------------------------------------------------------------------
extern "C" void kernel_launch(void* const* d_in, const int* in_sizes, int n_in,
                              void* d_out, int out_size, void* d_ws, size_t ws_size,
                              hipStream_t stream) {
  (void)in_sizes; (void)n_in; (void)out_size; (void)ws_size;
  const float* xyz = (const float*)d_in[0];
  const float* feat = (const float*)d_in[1];
  const float* W1 = (const float*)d_in[2];
  const float* g1 = (const float*)d_in[3];
  const float* b1 = (const float*)d_in[4];
  const float* W2 = (const float*)d_in[5];
  const float* g2 = (const float*)d_in[6];
  const float* b2 = (const float*)d_in[7];
  const float* W3 = (const float*)d_in[8];
  const float* g3 = (const float*)d_in[9];
  const float* b3 = (const float*)d_in[10];

  float* out = (float*)d_out;
  float* centroids = out;                       // (B,M,3)
  float* xout = out + (size_t)BB * MM * 3;      // (B,M,128)

  char* ws = (char*)d_ws;
  int* idx = (int*)(ws + 0);                    // B*M ints        (64 KB)
  int* gidx = (int*)(ws + (1 << 16));           // B*M*K ints      (2 MB)
  float* gsum1 = (float*)(ws + 2162688);        // BN stats (512 floats)
  float* gsq1 = gsum1 + 64;
  float* gsum2 = gsq1 + 64;
  float* gsq2 = gsum2 + 64;
  float* gsum3 = gsq2 + 64;
  float* gsq3 = gsum3 + 128;
  const size_t Y1off = (size_t)4 * 1024 * 1024;
  const size_t Ybytes64 = (size_t)NROWS * 64 * 4;   // 134 MB
  float* Y1 = (float*)(ws + Y1off);
  float* Y2 = (float*)(ws + Y1off + Ybytes64);
  float* Y3 = (float*)(ws + Y1off + 2 * Ybytes64);                   // 268 MB
  float* psum = (float*)(ws + Y1off + 2 * Ybytes64 + (size_t)NROWS * 128 * 4);
  float* psq = psum + (size_t)BB * MM * 128;

  const int nbm = BB * MM;  // 16384

  fps_kernel<<<BB, 256, 0, stream>>>(xyz, idx, centroids);
  knn_kernel<<<nbm, 256, 0, stream>>>(xyz, idx, gidx);

  gemm1_kernel<<<nbm, 256, 0, stream>>>(xyz, feat, centroids, gidx, W1, Y1, psum, psq);
  reduce_kernel<<<64, 256, 0, stream>>>(psum, gsum1, nbm, 64);
  reduce_kernel<<<64, 256, 0, stream>>>(psq, gsq1, nbm, 64);

  gemm2_kernel<<<nbm, 256, 0, stream>>>(Y1, gsum1, gsq1, g1, b1, W2, Y2, psum, psq);
  reduce_kernel<<<64, 256, 0, stream>>>(psum, gsum2, nbm, 64);
  reduce_kernel<<<64, 256, 0, stream>>>(psq, gsq2, nbm, 64);

  gemm3_kernel<<<nbm, 256, 0, stream>>>(Y2, gsum2, gsq2, g2, b2, W3, Y3, psum, psq);
  reduce_kernel<<<128, 256, 0, stream>>>(psum, gsum3, nbm, 128);
  reduce_kernel<<<128, 256, 0, stream>>>(psq, gsq3, nbm, 128);

  lse_kernel<<<nbm, 128, 0, stream>>>(Y3, gsum3, gsq3, g3, b3, xout);
}